// TFN_67920612819047
// MI455X (gfx1250) — compile-verified
//
#include <hip/hip_runtime.h>

// Trilinear tensor fusion: out[p, ((i*33)+j)*33 + k] = v_i * a_j * d_k
// where v/a/d are the 1-padded 33-vectors of x_vel/x_acc/x_2d at pair p.
//
// Store-bandwidth-bound (460 MB out @ 23.3 TB/s ~= 20 us floor).
// Uses V_WMMA_F32_16X16X4_F32 as a rank-1 outer-product engine:
// A = 16x4 fp32 with only the K=0 column holding va[M0..M0+15],
// B = 4x16 fp32 with only the K=0 row holding d[N0..N0+15],
// D[m][n] = va[M0+m] * d[N0+n]  (exact fp32, matches reference).

typedef __attribute__((ext_vector_type(2))) float v2f;
typedef __attribute__((ext_vector_type(8))) float v8f;

#define SEQ 50
#define M1 32
#define PDIM 33            // padded feature dim (1 + 32)
#define P2 1089            // 33*33 rows of the (i,j) x k view
#define PROWS_PAD 1104     // 69*16, zero-padded row table
#define DPAD 48            // 3*16, zero-padded d table
#define OUT_PER_PAIR 35937 // 33^3
#define MTILES 69          // ceil(1089/16)
#define NTILES 3           // ceil(33/16)
#define TILES (MTILES * NTILES)

__global__ __launch_bounds__(256) void trilinear_fusion_wmma(
    const float* __restrict__ xv,
    const float* __restrict__ xa,
    const float* __restrict__ xd,
    float* __restrict__ out)
{
    __shared__ float sV[PDIM];
    __shared__ float sA[PDIM];
    __shared__ float sD[DPAD];
    __shared__ float sVA[PROWS_PAD];

    const int p    = blockIdx.x;       // b*SEQ + t
    const int tid  = threadIdx.x;
    const int lane = tid & 31;         // wave32
    const int wave = tid >> 5;         // 8 waves / block

    // Stage padded v, a, d into LDS (leading implicit 1.0).
    if (tid < PDIM) {
        sV[tid] = (tid == 0) ? 1.0f : xv[p * M1 + tid - 1];
        sA[tid] = (tid == 0) ? 1.0f : xa[p * M1 + tid - 1];
        sD[tid] = (tid == 0) ? 1.0f : xd[p * M1 + tid - 1];
    } else if (tid < DPAD) {
        sD[tid] = 0.0f;                // zero-pad d up to 48
    }
    __syncthreads();

    // va[i*33+j] = v[i]*a[j], zero-padded to 1104 so 16-row tiles are safe.
    for (int n = tid; n < PROWS_PAD; n += 256) {
        sVA[n] = (n < P2) ? sV[n / PDIM] * sA[n % PDIM] : 0.0f;
    }
    __syncthreads();

    float* outBase = out + (size_t)p * OUT_PER_PAIR;
    const int   lo      = lane & 15;
    const float laneSel = (lane < 16) ? 1.0f : 0.0f;  // K=0 operands live in lanes 0..15

    // 207 tiles of 16x16 outputs, round-robin over the 8 waves.
    for (int tile = wave; tile < TILES; tile += 8) {
        const int mt = tile / NTILES;
        const int nt = tile - mt * NTILES;
        const int M0 = mt * 16;
        const int N0 = nt * 16;

        // A: 16x4 f32, lanes 0-15 VGPR0 = K=0 column; all other slots zero.
        v2f A;
        A.x = sVA[M0 + lo] * laneSel;
        A.y = 0.0f;
        // B: 4x16 f32, lanes 0-15 VGPR0 = K=0 row; all other slots zero.
        v2f Bm;
        Bm.x = sD[N0 + lo] * laneSel;
        Bm.y = 0.0f;

        v8f c = {};
        // (neg_a, A, neg_b, B, c_mod, C, reuse_a, reuse_b)
        c = __builtin_amdgcn_wmma_f32_16x16x4_f32(
                false, A, false, Bm, (short)0, c, false, false);

        // C/D layout: VGPR r, lanes 0-15 -> M = M0+r, lanes 16-31 -> M = M0+r+8.
        const int mBase = M0 + ((lane >> 4) << 3);
        const int N     = N0 + lo;

        if (mt < (MTILES - 1) && nt < 2) {
            // Full interior tile: unpredicated streaming stores
            // (each VGPR store = two contiguous 64B segments).
#pragma unroll
            for (int r = 0; r < 8; ++r) {
                __builtin_nontemporal_store(
                    c[r], &outBase[(size_t)(mBase + r) * PDIM + N]);
            }
        } else {
            // Edge tile: predicate per-lane against the real 1089x33 extent.
#pragma unroll
            for (int r = 0; r < 8; ++r) {
                const int M = mBase + r;
                if (M < P2 && N < PDIM) {
                    __builtin_nontemporal_store(
                        c[r], &outBase[(size_t)M * PDIM + N]);
                }
            }
        }
    }
}

extern "C" void kernel_launch(void* const* d_in, const int* in_sizes, int n_in,
                              void* d_out, int out_size, void* d_ws, size_t ws_size,
                              hipStream_t stream) {
    const float* xv = (const float*)d_in[0];
    const float* xa = (const float*)d_in[1];
    const float* xd = (const float*)d_in[2];
    float* out = (float*)d_out;

    const int B = in_sizes[0] / (SEQ * M1);   // 64 for the reference setup
    const int pairs = B * SEQ;                // 3200 workgroups

    trilinear_fusion_wmma<<<dim3(pairs), dim3(256), 0, stream>>>(xv, xa, xd, out);
}